// CustomCrossEntropyLoss_32727650795653
// MI455X (gfx1250) — compile-verified
//
#include <hip/hip_runtime.h>

// ---------------------------------------------------------------------------
// Weighted cross-entropy loss (N=4M rows, C=9 classes) -> two scalars.
// Memory-bound: ~160MB / 23.3 TB/s ~= 7us floor. Strategy: async-DMA 36KB
// tiles of logits into LDS via GLOBAL_LOAD_ASYNC_TO_LDS_B128 (gfx1250 async
// path, ASYNCcnt), log-softmax per row from LDS, wave32 butterfly + f32
// global atomics for the two running sums.
// ---------------------------------------------------------------------------

#define BLOCK 256
#define ROWS_PER_THREAD 4
#define TILE_ROWS (BLOCK * ROWS_PER_THREAD)      // 1024 rows / tile
#define NCLASS 9
#define TILE_DWORDS (TILE_ROWS * NCLASS)         // 9216 dwords = 36 KB
#define TILE_VEC4 (TILE_DWORDS / 4)              // 2304 float4
#define VEC4_PER_THREAD (TILE_VEC4 / BLOCK)      // 9 b128 per thread

#if defined(__has_builtin)
#if __has_builtin(__builtin_amdgcn_global_load_async_to_lds_b128)
#define CE_ASYNC_B128 1
#endif
#if __has_builtin(__builtin_amdgcn_global_load_async_to_lds_b32)
#define CE_ASYNC_B32 1
#endif
#endif

typedef int v4i __attribute__((vector_size(4 * sizeof(int))));
typedef __attribute__((address_space(1))) int* gint_ptr;
typedef __attribute__((address_space(3))) int* lint_ptr;
typedef __attribute__((address_space(1))) v4i* gv4i_ptr;
typedef __attribute__((address_space(3))) v4i* lv4i_ptr;

__device__ __forceinline__ void ce_wait_async_and_barrier() {
#if defined(CE_ASYNC_B128) || defined(CE_ASYNC_B32)
#if __has_builtin(__builtin_amdgcn_s_wait_asynccnt)
    __builtin_amdgcn_s_wait_asynccnt(0);
#else
    asm volatile("s_wait_asynccnt 0" ::: "memory");
#endif
#endif
    __syncthreads();
}

// Copy one float4 (global -> LDS), async if the gfx1250 path is available.
__device__ __forceinline__ void ce_copy16(const float* __restrict__ g, float* l) {
#if defined(CE_ASYNC_B128)
    __builtin_amdgcn_global_load_async_to_lds_b128((gv4i_ptr)g, (lv4i_ptr)l, 0, 0);
#elif defined(CE_ASYNC_B32)
#pragma unroll
    for (int i = 0; i < 4; ++i)
        __builtin_amdgcn_global_load_async_to_lds_b32((gint_ptr)(g + i), (lint_ptr)(l + i), 0, 0);
#else
    *(float4*)l = *(const float4*)g;
#endif
}

__device__ __forceinline__ void ce_copy4(const float* __restrict__ g, float* l) {
#if defined(CE_ASYNC_B32)
    __builtin_amdgcn_global_load_async_to_lds_b32((gint_ptr)g, (lint_ptr)l, 0, 0);
#else
    *l = *g;
#endif
}

__global__ __launch_bounds__(BLOCK) void ce_partial_kernel(
    const float* __restrict__ logits,   // [N, 9]
    const int* __restrict__ target,     // [N]
    const float* __restrict__ cweights, // [9]
    float* __restrict__ ws,             // ws[0]=sum(loss), ws[1]=count(loss>1e-16)
    int N) {
    __shared__ __align__(16) float s_log[TILE_DWORDS]; // 36 KB logits tile
    __shared__ float s_w[NCLASS];
    __shared__ float s_red[2 * (BLOCK / 32)];

    const int tid = threadIdx.x;
    if (tid < NCLASS) s_w[tid] = cweights[tid];

    const int numTiles = (N + TILE_ROWS - 1) / TILE_ROWS;
    float lsum = 0.0f;
    float lcnt = 0.0f;

    for (int tile = blockIdx.x; tile < numTiles; tile += gridDim.x) {
        const int rowBase = tile * TILE_ROWS;
        const int remRows = N - rowBase;
        const float* gsrc = logits + (long)rowBase * NCLASS;

        if (remRows >= TILE_ROWS) {
            // ---- full tile: unconditional 9x async b128 per thread ----
#pragma unroll
            for (int k = 0; k < VEC4_PER_THREAD; ++k) {
                const int v4 = tid + k * BLOCK;          // coalesced float4 index
                ce_copy16(gsrc + v4 * 4, &s_log[v4 * 4]);
            }
        } else {
            // ---- tail tile (at most one in the whole grid): b32 granularity ----
            const int tileDwords = remRows * NCLASS;
#pragma unroll
            for (int k = 0; k < NCLASS * ROWS_PER_THREAD; ++k) {
                const int idx = tid + k * BLOCK;
                if (idx < tileDwords) ce_copy4(gsrc + idx, &s_log[idx]);
            }
        }
        ce_wait_async_and_barrier();

        // ---- per-row log-softmax + weighted NLL from LDS ----
#pragma unroll
        for (int j = 0; j < ROWS_PER_THREAD; ++j) {
            const int lr = j * BLOCK + tid;              // local row in tile
            const int r = rowBase + lr;
            if (r < N) {
                const float* xr = &s_log[lr * NCLASS];   // stride 9: conflict-free
                float m = xr[0];
#pragma unroll
                for (int i = 1; i < NCLASS; ++i) m = fmaxf(m, xr[i]);
                float se = 0.0f;
#pragma unroll
                for (int i = 0; i < NCLASS; ++i) se += __expf(xr[i] - m);
                const float lse = m + __logf(se);
                const int t = target[r];                 // 0..8
                const float loss = s_w[t] * (lse - xr[t]);
                lsum += loss;
                if (loss > 1e-16f) lcnt += 1.0f;
            }
        }
        __syncthreads(); // protect LDS tile before next iteration's staging
    }

    // ---- wave32 butterfly reduction ----
#pragma unroll
    for (int off = 16; off > 0; off >>= 1) {
        lsum += __shfl_xor(lsum, off, 32);
        lcnt += __shfl_xor(lcnt, off, 32);
    }
    const int wave = tid >> 5;
    if ((tid & 31) == 0) {
        s_red[wave] = lsum;
        s_red[(BLOCK / 32) + wave] = lcnt;
    }
    __syncthreads();
    if (tid == 0) {
        float bs = 0.0f, bc = 0.0f;
#pragma unroll
        for (int w = 0; w < BLOCK / 32; ++w) {
            bs += s_red[w];
            bc += s_red[(BLOCK / 32) + w];
        }
        atomicAdd(&ws[0], bs);
        atomicAdd(&ws[1], bc);
    }
}

__global__ void ce_init_ws_kernel(float* __restrict__ ws) {
    ws[0] = 0.0f;
    ws[1] = 0.0f;
}

__global__ void ce_finalize_kernel(const float* __restrict__ ws,
                                   float* __restrict__ out, float invN) {
    const float s = ws[0];
    const float c = ws[1];
    out[0] = s / (c + 1e-16f); // loss_nonzero_mean
    out[1] = s * invN;         // losses.mean()
}

extern "C" void kernel_launch(void* const* d_in, const int* in_sizes, int n_in,
                              void* d_out, int out_size, void* d_ws, size_t ws_size,
                              hipStream_t stream) {
    const float* logits = (const float*)d_in[0];
    const int* target = (const int*)d_in[1];
    const float* cweights = (const float*)d_in[2];
    float* out = (float*)d_out;
    float* ws = (float*)d_ws;

    const int N = in_sizes[1]; // number of rows (target count)
    const int numTiles = (N + TILE_ROWS - 1) / TILE_ROWS;
    int blocks = numTiles < 2048 ? numTiles : 2048;
    if (blocks < 1) blocks = 1;

    ce_init_ws_kernel<<<1, 1, 0, stream>>>(ws);
    ce_partial_kernel<<<blocks, BLOCK, 0, stream>>>(logits, target, cweights, ws, N);
    ce_finalize_kernel<<<1, 1, 0, stream>>>(ws, out, N > 0 ? 1.0f / (float)N : 0.0f);
}